// DependencyBLSTM_80925773791577
// MI455X (gfx1250) — compile-verified
//
#include <hip/hip_runtime.h>
#include <math.h>

typedef __attribute__((ext_vector_type(2))) float v2f;
typedef __attribute__((ext_vector_type(8))) float v8f;

#define Dn 64
#define Sn 64
#define Ln 64
#define Wn 300
#define Hn 300
#define Mn (Dn * Sn)          // 4096 rows
#define NEGV (-9999999.0f)

// ---------------------------------------------------------------------------
// WMMA f32 16x16x4: one wave computes a 16x64 tile (4 accumulators).
// A-frag (16x4): lane l -> (M = l&15, K = kb, kb+1), kb = k0 + 2*(l>>4)
// B-frag (4x16): lane l -> (N = l&15, K = kb, kb+1)
// C/D (16x16):   lane l -> col (l&15), rows r + 8*(l>>4), r = 0..7
// Edge columns (n >= 300) are handled by clamping the B row to 299 (in-bounds
// duplicate data); their output columns are simply never stored. This keeps
// the K-loop control flow uniform -> no exec save/restore in the hot loop.
// ---------------------------------------------------------------------------
__device__ __forceinline__ v8f wmma_f32(v2f a, v2f b, v8f c) {
  return __builtin_amdgcn_wmma_f32_16x16x4_f32(false, a, false, b, (short)0, c,
                                               false, false);
}

__device__ __forceinline__ float lrelu(float x) {
  return x > 0.0f ? x : 0.01f * x;
}

// enc = lrelu( E[gather(idx)] @ W_sent^T + b_sent )   (4096 x 300, K = 300)
__global__ void k_enc(const int* __restrict__ idx, const float* __restrict__ E,
                      const float* __restrict__ Wm, const float* __restrict__ bias,
                      float* __restrict__ out) {
  const int lane = threadIdx.x;
  const int row = blockIdx.y * 16 + (lane & 15);
  const int n0  = blockIdx.x * 64;
  const int kh  = (lane >> 4) * 2;
  const float* arow = E + (size_t)idx[row * Ln + (Ln - 1)] * Wn + kh;
  const float* brow[4];
  int ncol[4];
#pragma unroll
  for (int t = 0; t < 4; ++t) {
    ncol[t] = n0 + t * 16 + (lane & 15);
    const int nb = ncol[t] < Hn ? ncol[t] : Hn - 1;
    brow[t] = Wm + (size_t)nb * Wn + kh;
  }
  v8f acc[4] = {{}, {}, {}, {}};
  for (int k0 = 0; k0 < Wn; k0 += 4) {
    v2f a = *(const v2f*)(arow + k0);
#pragma unroll
    for (int t = 0; t < 4; ++t)
      acc[t] = wmma_f32(a, *(const v2f*)(brow[t] + k0), acc[t]);
  }
  const int mbase = blockIdx.y * 16 + ((lane >> 4) ? 8 : 0);
#pragma unroll
  for (int t = 0; t < 4; ++t)
    if (ncol[t] < Hn) {
      const float bv = bias[ncol[t]];
      for (int r = 0; r < 8; ++r)
        out[(size_t)(mbase + r) * Hn + ncol[t]] = lrelu(acc[t][r] + bv);
    }
}

// Y = lrelu( X @ Wm^T + bias )   X: 4096 x 300 row-major, Wm: 300 x 300
__global__ void k_gemm_lrelu(const float* __restrict__ X, const float* __restrict__ Wm,
                             const float* __restrict__ bias, float* __restrict__ out) {
  const int lane = threadIdx.x;
  const int row = blockIdx.y * 16 + (lane & 15);
  const int n0  = blockIdx.x * 64;
  const int kh  = (lane >> 4) * 2;
  const float* arow = X + (size_t)row * Hn + kh;
  const float* brow[4];
  int ncol[4];
#pragma unroll
  for (int t = 0; t < 4; ++t) {
    ncol[t] = n0 + t * 16 + (lane & 15);
    const int nb = ncol[t] < Hn ? ncol[t] : Hn - 1;
    brow[t] = Wm + (size_t)nb * Hn + kh;
  }
  v8f acc[4] = {{}, {}, {}, {}};
  for (int k0 = 0; k0 < Hn; k0 += 4) {
    v2f a = *(const v2f*)(arow + k0);
#pragma unroll
    for (int t = 0; t < 4; ++t)
      acc[t] = wmma_f32(a, *(const v2f*)(brow[t] + k0), acc[t]);
  }
  const int mbase = blockIdx.y * 16 + ((lane >> 4) ? 8 : 0);
#pragma unroll
  for (int t = 0; t < 4; ++t)
    if (ncol[t] < Hn) {
      const float bv = bias[ncol[t]];
      for (int r = 0; r < 8; ++r)
        out[(size_t)(mbase + r) * Hn + ncol[t]] = lrelu(acc[t][r] + bv);
    }
}

// A_raw[d,i,j] = sum_h P[d,i,h] * C[d,j,h]   (per-doc 64x64, K = 300)
// One wave per 16x64 tile: grid (itile=4, d=64).
__global__ void k_batched_nt(const float* __restrict__ P, const float* __restrict__ C,
                             float* __restrict__ Araw) {
  const int lane = threadIdx.x;
  const int d = blockIdx.y;
  const int i = blockIdx.x * 16 + (lane & 15);
  const int kh = (lane >> 4) * 2;
  const float* arow = P + (size_t)(d * Sn + i) * Hn + kh;
  const float* brow[4];
#pragma unroll
  for (int t = 0; t < 4; ++t)
    brow[t] = C + (size_t)(d * Sn + t * 16 + (lane & 15)) * Hn + kh;
  v8f acc[4] = {{}, {}, {}, {}};
  for (int k0 = 0; k0 < Hn; k0 += 4) {
    v2f a = *(const v2f*)(arow + k0);
#pragma unroll
    for (int t = 0; t < 4; ++t)
      acc[t] = wmma_f32(a, *(const v2f*)(brow[t] + k0), acc[t]);
  }
  const int mbase = blockIdx.x * 16 + ((lane >> 4) ? 8 : 0);
  float* dst = Araw + (size_t)d * Sn * Sn;
#pragma unroll
  for (int t = 0; t < 4; ++t) {
    const int j = t * 16 + (lane & 15);
    for (int r = 0; r < 8; ++r)
      dst[(mbase + r) * Sn + j] = acc[t][r];
  }
}

// Column softmax over i with diagonal masked to NEG (exp underflows to 0).
__global__ void k_softmaxA(const float* __restrict__ Araw, float* __restrict__ outA) {
  const int d = blockIdx.x, j = threadIdx.x;
  const float* col = Araw + (size_t)d * Sn * Sn;
  float mx = NEGV;
  for (int i = 0; i < Sn; ++i) {
    float v = (i == j) ? NEGV : col[i * Sn + j];
    mx = fmaxf(mx, v);
  }
  float sum = 0.0f;
  for (int i = 0; i < Sn; ++i) {
    float v = (i == j) ? NEGV : col[i * Sn + j];
    sum += expf(v - mx);
  }
  const float inv = 1.0f / sum;
  float* dst = outA + (size_t)d * Sn * Sn;
  for (int i = 0; i < Sn; ++i) {
    float v = (i == j) ? NEGV : col[i * Sn + j];
    dst[i * Sn + j] = expf(v - mx) * inv;
  }
}

// rowsum[d,i] = sum_j A[d,i,j]
__global__ void k_rowsum(const float* __restrict__ A, float* __restrict__ rs) {
  const int d = blockIdx.x, i = threadIdx.x;
  const float* row = A + (size_t)d * Sn * Sn + (size_t)i * Sn;
  float s = 0.0f;
  for (int j = 0; j < Sn; ++j) s += row[j];
  rs[d * Sn + i] = s;
}

// fri = softmax_s( enc[d,s,:] . w_root + b_root )
__global__ void k_fri(const float* __restrict__ enc, const float* __restrict__ w_root,
                      const float* __restrict__ b_root, float* __restrict__ fri) {
  __shared__ float sm[Sn];
  __shared__ float red;
  const int d = blockIdx.x, s = threadIdx.x;
  const float* row = enc + (size_t)(d * Sn + s) * Hn;
  float acc = b_root[0];
  for (int k = 0; k < Hn; ++k) acc += row[k] * w_root[k];
  sm[s] = acc;
  __syncthreads();
  if (s == 0) {
    float m = sm[0];
    for (int t = 1; t < Sn; ++t) m = fmaxf(m, sm[t]);
    red = m;
  }
  __syncthreads();
  const float e = expf(acc - red);
  sm[s] = e;
  __syncthreads();
  if (s == 0) {
    float sum = 0.0f;
    for (int t = 0; t < Sn; ++t) sum += sm[t];
    red = 1.0f / sum;
  }
  __syncthreads();
  fri[d * Sn + s] = e * red;
}

// agg[d,i,h] = sum_k A[d,k,i]*enc[d,k,h] + fri[d,i]*root_embed[h]   (K = 64)
// One wave per 16x64 tile: grid (ntile=5, itile=4, d=64).
__global__ void k_aggT(const float* __restrict__ A, const float* __restrict__ enc,
                       const float* __restrict__ fri, const float* __restrict__ root_embed,
                       float* __restrict__ agg) {
  const int lane = threadIdx.x;
  const int d = blockIdx.z;
  const int i = blockIdx.y * 16 + (lane & 15);
  const int n0 = blockIdx.x * 64;
  const int kh = (lane >> 4) * 2;
  const float* Ad = A + (size_t)d * Sn * Sn;
  const float* encd = enc + (size_t)d * Sn * Hn;
  int ncol[4], nb[4];
#pragma unroll
  for (int t = 0; t < 4; ++t) {
    ncol[t] = n0 + t * 16 + (lane & 15);
    nb[t] = ncol[t] < Hn ? ncol[t] : Hn - 1;
  }
  v8f acc[4] = {{}, {}, {}, {}};
  for (int k0 = 0; k0 < Sn; k0 += 4) {
    const int kb = k0 + kh;
    v2f a;
    a.x = Ad[kb * Sn + i];
    a.y = Ad[(kb + 1) * Sn + i];
#pragma unroll
    for (int t = 0; t < 4; ++t) {
      v2f b;
      b.x = encd[(size_t)kb * Hn + nb[t]];
      b.y = encd[(size_t)(kb + 1) * Hn + nb[t]];
      acc[t] = wmma_f32(a, b, acc[t]);
    }
  }
  const int mbase = blockIdx.y * 16 + ((lane >> 4) ? 8 : 0);
#pragma unroll
  for (int t = 0; t < 4; ++t)
    if (ncol[t] < Hn) {
      const float re = root_embed[ncol[t]];
      for (int r = 0; r < 8; ++r) {
        const int li = mbase + r;
        agg[(size_t)(d * Sn + li) * Hn + ncol[t]] = acc[t][r] + fri[d * Sn + li] * re;
      }
    }
}

// ri = lrelu( [enc | agg | enc*rowsum] @ W_r^T + b_r )   (4096 x 900 -> 300)
// Three branchless K-segment loops (segment bounds are multiples of 4).
__global__ void k_ri(const float* __restrict__ enc, const float* __restrict__ agg,
                     const float* __restrict__ rs, const float* __restrict__ W_r,
                     const float* __restrict__ b_r, float* __restrict__ ri) {
  const int lane = threadIdx.x;
  const int row = blockIdx.y * 16 + (lane & 15);
  const int n0  = blockIdx.x * 64;
  const int kh  = (lane >> 4) * 2;
  const float* arow1 = enc + (size_t)row * Hn + kh;
  const float* arow2 = agg + (size_t)row * Hn + kh;
  const float  rsv   = rs[row];
  const float* brow[4];
  int ncol[4];
#pragma unroll
  for (int t = 0; t < 4; ++t) {
    ncol[t] = n0 + t * 16 + (lane & 15);
    const int nb = ncol[t] < Hn ? ncol[t] : Hn - 1;
    brow[t] = W_r + (size_t)nb * (3 * Hn) + kh;
  }
  v8f acc[4] = {{}, {}, {}, {}};
  for (int k0 = 0; k0 < Hn; k0 += 4) {          // segment 1: enc
    v2f a = *(const v2f*)(arow1 + k0);
#pragma unroll
    for (int t = 0; t < 4; ++t)
      acc[t] = wmma_f32(a, *(const v2f*)(brow[t] + k0), acc[t]);
  }
  for (int k0 = 0; k0 < Hn; k0 += 4) {          // segment 2: agg
    v2f a = *(const v2f*)(arow2 + k0);
#pragma unroll
    for (int t = 0; t < 4; ++t)
      acc[t] = wmma_f32(a, *(const v2f*)(brow[t] + Hn + k0), acc[t]);
  }
  for (int k0 = 0; k0 < Hn; k0 += 4) {          // segment 3: enc * rowsum
    v2f a = *(const v2f*)(arow1 + k0);
    a.x *= rsv;
    a.y *= rsv;
#pragma unroll
    for (int t = 0; t < 4; ++t)
      acc[t] = wmma_f32(a, *(const v2f*)(brow[t] + 2 * Hn + k0), acc[t]);
  }
  const int mbase = blockIdx.y * 16 + ((lane >> 4) ? 8 : 0);
#pragma unroll
  for (int t = 0; t < 4; ++t)
    if (ncol[t] < Hn) {
      const float bv = b_r[ncol[t]];
      for (int r = 0; r < 8; ++r)
        ri[(size_t)(mbase + r) * Hn + ncol[t]] = lrelu(acc[t][r] + bv);
    }
}

// final[d,h] = mean_s ri[d,s,h]
__global__ void k_mean(const float* __restrict__ ri, float* __restrict__ fin) {
  const int d = blockIdx.x, h = threadIdx.x;
  if (h >= Hn) return;
  float s = 0.0f;
  for (int t = 0; t < Sn; ++t) s += ri[(size_t)(d * Sn + t) * Hn + h];
  fin[d * Hn + h] = s * (1.0f / Sn);
}

// out[d,c] = final[d,:] . W_cls[c,:] + b_cls[c]
__global__ void k_out(const float* __restrict__ fin, const float* __restrict__ W_cls,
                      const float* __restrict__ b_cls, float* __restrict__ out) {
  __shared__ float sm[64];
  const int c = blockIdx.x & 1, d = blockIdx.x >> 1, t = threadIdx.x;
  float p = 0.0f;
  for (int k = t; k < Hn; k += 64) p += fin[d * Hn + k] * W_cls[c * Hn + k];
  sm[t] = p;
  __syncthreads();
  if (t == 0) {
    float s = 0.0f;
    for (int q = 0; q < 64; ++q) s += sm[q];
    out[d * 2 + c] = s + b_cls[c];
  }
}

extern "C" void kernel_launch(void* const* d_in, const int* in_sizes, int n_in,
                              void* d_out, int out_size, void* d_ws, size_t ws_size,
                              hipStream_t stream) {
  const int*   wi       = (const int*)d_in[0];
  const float* E        = (const float*)d_in[1];
  const float* W_sent   = (const float*)d_in[2];
  const float* b_sent   = (const float*)d_in[3];
  const float* W_par    = (const float*)d_in[4];
  const float* b_par    = (const float*)d_in[5];
  const float* W_ch     = (const float*)d_in[6];
  const float* b_ch     = (const float*)d_in[7];
  const float* w_root   = (const float*)d_in[8];
  const float* b_root   = (const float*)d_in[9];
  const float* root_emb = (const float*)d_in[10];
  const float* W_r      = (const float*)d_in[11];
  const float* b_r      = (const float*)d_in[12];
  const float* W_cls    = (const float*)d_in[13];
  const float* b_cls    = (const float*)d_in[14];

  float* out    = (float*)d_out;                 // (64,2) = 128 floats
  float* outA   = out + 128;                     // (64,64,64) = 262144
  float* outFri = outA + Dn * Sn * Sn;           // (64,64) = 4096

  float* ws   = (float*)d_ws;
  float* enc  = ws;                              // 4096*300
  float* P    = enc  + (size_t)Mn * Hn;
  float* C    = P    + (size_t)Mn * Hn;
  float* Araw = C    + (size_t)Mn * Hn;          // 64*64*64
  float* agg  = Araw + (size_t)Dn * Sn * Sn;     // 4096*300
  float* ri   = agg  + (size_t)Mn * Hn;          // 4096*300
  float* rsum = ri   + (size_t)Mn * Hn;          // 4096
  float* fin  = rsum + Mn;                       // 64*300

  const dim3 gBig((Hn + 63) / 64, Mn / 16);      // (5, 256)
  const dim3 wave(32);

  k_enc<<<gBig, wave, 0, stream>>>(wi, E, W_sent, b_sent, enc);
  k_gemm_lrelu<<<gBig, wave, 0, stream>>>(enc, W_par, b_par, P);
  k_gemm_lrelu<<<gBig, wave, 0, stream>>>(enc, W_ch, b_ch, C);
  k_batched_nt<<<dim3(Sn / 16, Dn), wave, 0, stream>>>(P, C, Araw);
  k_softmaxA<<<Dn, Sn, 0, stream>>>(Araw, outA);
  k_rowsum<<<Dn, Sn, 0, stream>>>(outA, rsum);
  k_fri<<<Dn, Sn, 0, stream>>>(enc, w_root, b_root, outFri);
  k_aggT<<<dim3((Hn + 63) / 64, Sn / 16, Dn), wave, 0, stream>>>(outA, enc, outFri,
                                                                 root_emb, agg);
  k_ri<<<gBig, wave, 0, stream>>>(enc, agg, rsum, W_r, b_r, ri);
  k_mean<<<Dn, 320, 0, stream>>>(ri, fin);
  k_out<<<Dn * 2, 64, 0, stream>>>(fin, W_cls, b_cls, out);
}